// LogSparsemaxBisect_12421045420949
// MI455X (gfx1250) — compile-verified
//
#include <hip/hip_runtime.h>
#include <math.h>

#define DCOL   32000
#define NT     1024
#define NWAVE  32
#define CAP    2048
#define NITER  50

#if __has_builtin(__builtin_amdgcn_global_load_async_to_lds_b128)
#define HAVE_ASYNC_LDS 1
#else
#define HAVE_ASYNC_LDS 0
#endif

typedef int   v4i __attribute__((vector_size(16)));
typedef float v4f __attribute__((ext_vector_type(4)));
typedef __attribute__((address_space(3))) v4i lds_v4i;

__device__ __forceinline__ float waveSum(float v) {
#pragma unroll
  for (int o = 16; o; o >>= 1) v += __shfl_xor(v, o, 32);
  return v;
}
__device__ __forceinline__ float waveMax(float v) {
#pragma unroll
  for (int o = 16; o; o >>= 1) v = fmaxf(v, __shfl_xor(v, o, 32));
  return v;
}

__global__ __launch_bounds__(NT)
void logsparsemax_bisect_kernel(const float* __restrict__ X, float* __restrict__ out) {
  // Row-resident LDS: 128000B row + 8KB candidates + reduction scratch (~136KB)
  __shared__ __attribute__((aligned(16))) float sh[DCOL];
  __shared__ float cand[CAP];
  __shared__ float red[NWAVE];
  __shared__ float s_b[4];   // [0]=rowmax [1]=tau_m [2]=sum_p [3]=tmp
  __shared__ int   s_cnt;

  const int tid  = threadIdx.x;
  const int lane = tid & 31;
  const int wid  = tid >> 5;
  const int row  = blockIdx.x;
  const float* rowp = X + (size_t)row * DCOL;
  constexpr int CH = DCOL / 4;          // 8000 x 16B chunks (exact, no tail)
  const v4f* sh4 = (const v4f*)sh;

  // ---------------- stage row into LDS (gfx1250 async-to-LDS path) -------------
#if HAVE_ASYNC_LDS
  {
    v4i* g = (v4i*)rowp;                // flat pointer to 16B chunks (drops const)
    lds_v4i* l = (lds_v4i*)sh;          // LDS (AS3) pointer to 16B chunks
    for (int c = tid; c < CH; c += NT) {
      __builtin_amdgcn_global_load_async_to_lds_b128(g + c, l + c, 0, 0);
    }
#if __has_builtin(__builtin_amdgcn_s_wait_asynccnt)
    __builtin_amdgcn_s_wait_asynccnt(0);
#else
    asm volatile("s_wait_asynccnt 0" ::: "memory");
#endif
  }
#else
  for (int c = tid; c < DCOL; c += NT) sh[c] = rowp[c];
#endif
  if (tid == 0) s_cnt = 0;
  __syncthreads();

  // ---------------- row max (b128 LDS reads, wave32 butterfly) -----------------
  float lm = -__builtin_inff();
  for (int c = tid; c < CH; c += NT) {
    v4f x = sh4[c];
    lm = fmaxf(lm, fmaxf(fmaxf(x.x, x.y), fmaxf(x.z, x.w)));
  }
  lm = waveMax(lm);
  if (lane == 0) red[wid] = lm;
  __syncthreads();
  if (wid == 0) {
    float v = waveMax(red[lane]);
    if (lane == 0) s_b[0] = v;
  }
  __syncthreads();
  const float rowmax = s_b[0];
  const float thr    = rowmax - 1.0f;                 // tau_lo(0); only x>thr can matter

  // ---------------- gather candidate set (tau never drops below thr) -----------
  for (int c = tid; c < CH; c += NT) {
    v4f x = sh4[c];
#pragma unroll
    for (int k = 0; k < 4; ++k) {
      float xv = x[k];
      if (xv > thr) {
        int idx = atomicAdd(&s_cnt, 1);               // ds_add_rtn_u32
        if (idx < CAP) cand[idx] = xv;
      }
    }
  }
  __syncthreads();
  const int   cnt = s_cnt;
  const float dm0 = (rowmax - (1.0f / (float)DCOL)) - thr;   // tau_hi - tau_lo (f32, as ref)

  if (cnt <= CAP) {
    // ---- fast path: wave 0 runs all 50 bisection iters on the tiny set, no barriers
    if (wid == 0) {
      float s = 0.0f;
      for (int i = lane; i < cnt; i += 32) s += cand[i] - thr;   // clip no-op: cand>thr
      const float f_lo = waveSum(s) - 1.0f;
      float tau_lo = thr, dm = dm0, tau_m = thr;
#pragma unroll 1
      for (int it = 0; it < NITER; ++it) {
        dm *= 0.5f;
        tau_m = tau_lo + dm;
        float fm = 0.0f;
        for (int i = lane; i < cnt; i += 32) fm += fmaxf(cand[i] - tau_m, 0.0f);
        fm = waveSum(fm) - 1.0f;
        if (fm * f_lo >= 0.0f) tau_lo = tau_m;                   // exact ref semantics
      }
      float sp = 0.0f;
      for (int i = lane; i < cnt; i += 32) sp += fmaxf(cand[i] - tau_m, 0.0f);
      sp = waveSum(sp);
      if (lane == 0) { s_b[1] = tau_m; s_b[2] = sp; }
    }
  } else {
    // ---- fallback: block-wide bisection over the full LDS-resident row --------
    float tau_lo = thr, dm = dm0, tau_m = thr;
    float s = 0.0f;
    for (int c = tid; c < DCOL; c += NT) s += fmaxf(sh[c] - thr, 0.0f);
    s = waveSum(s);
    if (lane == 0) red[wid] = s;
    __syncthreads();
    if (wid == 0) {
      float v = waveSum(red[lane]);
      if (lane == 0) s_b[3] = v;
    }
    __syncthreads();
    const float f_lo = s_b[3] - 1.0f;
#pragma unroll 1
    for (int it = 0; it < NITER; ++it) {
      dm *= 0.5f;
      tau_m = tau_lo + dm;
      float fm = 0.0f;
      for (int c = tid; c < DCOL; c += NT) fm += fmaxf(sh[c] - tau_m, 0.0f);
      fm = waveSum(fm);
      __syncthreads();
      if (lane == 0) red[wid] = fm;
      __syncthreads();
      if (wid == 0) {
        float v = waveSum(red[lane]);
        if (lane == 0) s_b[3] = v;
      }
      __syncthreads();
      fm = s_b[3] - 1.0f;
      if (fm * f_lo >= 0.0f) tau_lo = tau_m;
    }
    float sp = 0.0f;
    for (int c = tid; c < DCOL; c += NT) sp += fmaxf(sh[c] - tau_m, 0.0f);
    sp = waveSum(sp);
    __syncthreads();
    if (lane == 0) red[wid] = sp;
    __syncthreads();
    if (wid == 0) {
      float v = waveSum(red[lane]);
      if (lane == 0) { s_b[1] = tau_m; s_b[2] = v; }
    }
  }
  __syncthreads();

  // ---------------- output: log(p / sum_p), -inf outside support ---------------
  const float tau = s_b[1];
  const float inv = 1.0f / s_b[2];
  v4f* out4 = (v4f*)(out + (size_t)row * DCOL);
  const float ninf = -__builtin_inff();
  for (int c = tid; c < CH; c += NT) {
    v4f x = sh4[c];
    v4f r;
#pragma unroll
    for (int k = 0; k < 4; ++k) {
      float xv = x[k];
      r[k] = (xv > tau) ? logf((xv - tau) * inv) : ninf;
    }
    __builtin_nontemporal_store(r, out4 + c);   // streamed write-once: NT hint
  }
}

extern "C" void kernel_launch(void* const* d_in, const int* in_sizes, int n_in,
                              void* d_out, int out_size, void* d_ws, size_t ws_size,
                              hipStream_t stream) {
  const float* X = (const float*)d_in[0];
  float* out = (float*)d_out;
  const int rows = in_sizes[0] / DCOL;   // 4096
  logsparsemax_bisect_kernel<<<rows, NT, 0, stream>>>(X, out);
}